// GaussianMomentsEmbedding_71794673320010
// MI455X (gfx1250) — compile-verified
//
#include <hip/hip_runtime.h>
#include <hip/hip_bf16.h>

#define NCHAN 7
#define RDIM 8
#define ADIM 16
#define NATOMS 30000
#define NEDGES 960000
#define NOUT 443
#define CUTF 5.2f

typedef __attribute__((ext_vector_type(2))) float v2f;
typedef __attribute__((ext_vector_type(8))) float v8f;

// ---------------- workspace layout (in floats) ----------------
// [0,896)      PStab[s][j][k] = sum_i species_table[s][i]*chem[i][j][k]
// [896,941)    W112 (3,3,5)
// [941,1066)   W222 (5,5,5)
// [1066,1171)  W123 (3,5,7)
// [1171,1416)  W233 (5,7,7)
// ints at float-offset 1416: PS[28] PD[28] T1[84] T2[84] T3[84]
// [2048, 2048+3360000)  rhoi[NATOMS][7][16]
#define WS_W112 896
#define WS_W222 941
#define WS_W123 1066
#define WS_W233 1171
#define WS_IL   1416
#define WS_RHOI 2048

// ---------------- device CG / real-SH change-of-basis ----------------
__device__ double dfact(int n) { double r = 1.0; for (int i = 2; i <= n; ++i) r *= (double)i; return r; }

__device__ double cg_coef(int j1, int m1, int j2, int m2, int j3, int m3) {
    if (m1 + m2 != m3) return 0.0;
    double pre = sqrt((2.0 * j3 + 1.0) * dfact(j3 + j1 - j2) * dfact(j3 - j1 + j2) *
                      dfact(j1 + j2 - j3) / dfact(j1 + j2 + j3 + 1));
    pre *= sqrt(dfact(j3 + m3) * dfact(j3 - m3) * dfact(j1 - m1) * dfact(j1 + m1) *
                dfact(j2 - m2) * dfact(j2 + m2));
    double s = 0.0;
    for (int k = 0; k <= j1 + j2 - j3; ++k) {
        int d0 = k, d1 = j1 + j2 - j3 - k, d2 = j1 - m1 - k;
        int d3 = j2 + m2 - k, d4 = j3 - j2 + m1 + k, d5 = j3 - j1 - m2 + k;
        if (d0 < 0 || d1 < 0 || d2 < 0 || d3 < 0 || d4 < 0 || d5 < 0) continue;
        double den = dfact(d0) * dfact(d1) * dfact(d2) * dfact(d3) * dfact(d4) * dfact(d5);
        s += ((k & 1) ? -1.0 : 1.0) / den;
    }
    return pre * s;
}

struct URow { int n; int col[2]; double re[2]; double im[2]; };
__device__ URow urow(int l, int a) {
    URow r; const double s2 = 0.70710678118654752440;
    int m = a - l;
    if (m == 0) { r.n = 1; r.col[0] = l; r.re[0] = 1.0; r.im[0] = 0.0; }
    else if (m > 0) {
        r.n = 2;
        r.col[0] = l + m; r.re[0] = ((m & 1) ? -s2 : s2); r.im[0] = 0.0;
        r.col[1] = l - m; r.re[1] = s2;                   r.im[1] = 0.0;
    } else {
        int mm = -m; r.n = 2;
        r.col[0] = l - mm; r.re[0] = 0.0; r.im[0] = s2;
        r.col[1] = l + mm; r.re[1] = 0.0; r.im[1] = ((mm & 1) ? s2 : -s2);
    }
    return r;
}

__device__ float wentry(int l1, int l2, int l3, int a, int b, int c) {
    URow u1 = urow(l1, a), u2 = urow(l2, b), u3 = urow(l3, c);
    double res = 0.0;
    for (int i = 0; i < u1.n; ++i)
        for (int j = 0; j < u2.n; ++j)
            for (int k = 0; k < u3.n; ++k) {
                double g = cg_coef(l1, u1.col[i] - l1, l2, u2.col[j] - l2, l3, u3.col[k] - l3);
                if (g == 0.0) continue;
                // p = U1 * U2 * conj(U3); only the real part matters
                double pr = u1.re[i] * u2.re[j] - u1.im[i] * u2.im[j];
                double pi = u1.re[i] * u2.im[j] + u1.im[i] * u2.re[j];
                double qr = pr * u3.re[k] + pi * u3.im[k];   // times conj -> re
                res += qr * g;
            }
    return (float)res;
}

// ---------------- kernel: build PStab, W tensors, index lists ----------------
__global__ void build_tables(const float* __restrict__ table, const float* __restrict__ chem,
                             float* __restrict__ ws, int* __restrict__ il) {
    int tid = threadIdx.x;
    // PStab: per-species pre-contraction of the 'ai,ijk' part of the einsum
    for (int idx = tid; idx < ADIM * RDIM * NCHAN; idx += 256) {
        int s = idx / 56, r = idx % 56, j = r / 7, k = r % 7;
        float acc = 0.f;
        for (int i = 0; i < ADIM; ++i) acc += table[s * ADIM + i] * chem[i * 56 + j * 7 + k];
        ws[idx] = acc;
    }
    for (int idx = tid; idx < 45;  idx += 256) ws[WS_W112 + idx] = wentry(1, 1, 2, idx / 15, (idx / 5) % 3, idx % 5);
    for (int idx = tid; idx < 125; idx += 256) ws[WS_W222 + idx] = wentry(2, 2, 2, idx / 25, (idx / 5) % 5, idx % 5);
    for (int idx = tid; idx < 105; idx += 256) ws[WS_W123 + idx] = wentry(1, 2, 3, idx / 35, (idx / 7) % 5, idx % 7);
    for (int idx = tid; idx < 245; idx += 256) ws[WS_W233 + idx] = wentry(2, 3, 3, idx / 49, (idx / 7) % 7, idx % 7);
    if (tid == 0) {
        int p = 0, t = 0;
        for (int i = 0; i < NCHAN; ++i)
            for (int j = i; j < NCHAN; ++j) {
                il[p] = i; il[28 + p] = j; ++p;
                for (int k = j; k < NCHAN; ++k) { il[56 + t] = i; il[140 + t] = j; il[224 + t] = k; ++t; }
            }
    }
}

__global__ void zero_kernel(float* __restrict__ p, int n) {
    int i = blockIdx.x * blockDim.x + threadIdx.x;
    if (i < n) p[i] = 0.f;
}

// ---------------- kernel: per-edge rho scatter ----------------
__global__ __launch_bounds__(256) void edge_kernel(
    const int* __restrict__ esrc, const int* __restrict__ edst,
    const float* __restrict__ dist, const float* __restrict__ vec,
    const float* __restrict__ sw, const int* __restrict__ species,
    const float* __restrict__ ws, float* __restrict__ rhoi) {
    int e = blockIdx.x * blockDim.x + threadIdx.x;
    if (e >= NEDGES) return;
    float d = dist[e];
    float s = sw[e];
    int a_src = esrc[e];
    int sp = species[edst[e]];
    const float* P = ws + sp * 56;

    // Bessel radial basis
    float invd = 1.0f / d;
    float t = d * (3.14159265358979323846f / CUTF);
    const float nrm = 0.62017367294604225f; // sqrt(2/5.2)
    float rad[RDIM];
#pragma unroll
    for (int n = 0; n < RDIM; ++n) rad[n] = nrm * __sinf((float)(n + 1) * t) * invd;

    // xij[k] = sum_j rad[j] * PStab[sp][j][k], scaled by switch
    float x[NCHAN] = {0.f, 0.f, 0.f, 0.f, 0.f, 0.f, 0.f};
#pragma unroll
    for (int j = 0; j < RDIM; ++j) {
        float rj = rad[j];
#pragma unroll
        for (int k = 0; k < NCHAN; ++k) x[k] += rj * P[j * 7 + k];
    }
#pragma unroll
    for (int k = 0; k < NCHAN; ++k) x[k] *= s;

    // real spherical harmonics (l = 0..3), reference ordering
    float vx = vec[3 * e + 0], vy = vec[3 * e + 1], vz = vec[3 * e + 2];
    float inv = 1.0f / sqrtf(vx * vx + vy * vy + vz * vz);
    float ux = vx * inv, uy = vy * inv, uz = vz * inv;
    float x2 = ux * ux, y2 = uy * uy, z2 = uz * uz;
    float Y[16];
    Y[0] = 0.28209479177387814f;
    const float c1 = 0.4886025119029199f;
    Y[1] = c1 * uy; Y[2] = c1 * uz; Y[3] = c1 * ux;
    Y[4] = 1.0925484305920792f * ux * uy;
    Y[5] = 1.0925484305920792f * uy * uz;
    Y[6] = 0.31539156525252005f * (3.f * z2 - 1.f);
    Y[7] = 1.0925484305920792f * ux * uz;
    Y[8] = 0.5462742152960396f * (x2 - y2);
    Y[9] = 0.5900435899266435f * uy * (3.f * x2 - y2);
    Y[10] = 2.890611442640554f * ux * uy * uz;
    Y[11] = 0.4570457994644658f * uy * (5.f * z2 - 1.f);
    Y[12] = 0.3731763325901154f * uz * (5.f * z2 - 3.f);
    Y[13] = 0.4570457994644658f * ux * (5.f * z2 - 1.f);
    Y[14] = 1.445305721320277f * uz * (x2 - y2);
    Y[15] = 0.5900435899266435f * ux * (3.f * x2 - y2);

    float* dp = rhoi + (size_t)a_src * 112;
#pragma unroll
    for (int c = 0; c < NCHAN; ++c) {
        float xc = x[c];
#pragma unroll
        for (int m = 0; m < 16; ++m) unsafeAtomicAdd(&dp[c * 16 + m], xc * Y[m]);
    }
}

// ---------------- kernel: one-hot embedding via WMMA f32 16x16x4 ----------------
// out[a, 0:16] = species_table[species[a]]  ==  indicator(30000x16) x table(16x16)
__global__ __launch_bounds__(256) void onehot_wmma(const int* __restrict__ species,
                                                   const float* __restrict__ table,
                                                   float* __restrict__ out) {
    int wave = (int)((blockIdx.x * blockDim.x + threadIdx.x) >> 5);
    int lane = threadIdx.x & 31;
    int base = wave * 16;
    if (base >= NATOMS) return;   // wave-uniform exit; EXEC all-ones where WMMA runs
    int M = lane & 15;            // A: M index; D/B: N index
    int half = lane >> 4;
    int sp = species[base + M];
    v8f c = {};
#pragma unroll
    for (int kk = 0; kk < 4; ++kk) {
        int k0 = kk * 4 + half * 2;
        v2f a, b;
        a.x = (sp == k0)     ? 1.0f : 0.0f;   // A[M][k0]
        a.y = (sp == k0 + 1) ? 1.0f : 0.0f;   // A[M][k0+1]
        b.x = table[k0 * 16 + M];             // B[k0][N]
        b.y = table[(k0 + 1) * 16 + M];       // B[k0+1][N]
        c = __builtin_amdgcn_wmma_f32_16x16x4_f32(false, a, false, b, (short)0, c,
                                                  false, false);
    }
#pragma unroll
    for (int r = 0; r < 8; ++r)
        out[(size_t)(base + r + 8 * half) * NOUT + M] = c[r];
}

// ---------------- kernel: per-atom invariant contractions ----------------
__global__ __launch_bounds__(256) void atom_kernel(const float* __restrict__ rhoi,
                                                   const float* __restrict__ ws,
                                                   const int* __restrict__ il,
                                                   float* __restrict__ out) {
    __shared__ float sr[8][112];
    int lane = threadIdx.x & 31;
    int w = threadIdx.x >> 5;
    int a = blockIdx.x * 8 + w;
    bool active = (a < NATOMS);
    if (active)
        for (int idx = lane; idx < 112; idx += 32) sr[w][idx] = rhoi[(size_t)a * 112 + idx];
    __syncthreads();
    if (!active) return;

    const float* W112 = ws + WS_W112;
    const float* W222 = ws + WS_W222;
    const float* W123 = ws + WS_W123;
    const float* W233 = ws + WS_W233;
    const int* PSl = il;       const int* PDl = il + 28;
    const int* T1l = il + 56;  const int* T2l = il + 140;  const int* T3l = il + 224;
    const float* R = sr[w];
    float* op = out + (size_t)a * NOUT;

    if (lane < NCHAN) op[16 + lane] = R[lane * 16];   // xi0

    for (int p = lane; p < 28; p += 32) {
        const float* Ri = R + PSl[p] * 16;
        const float* Rj = R + PDl[p] * 16;
        float s1 = 0.f, s2 = 0.f, s3 = 0.f;
        for (int m = 1; m < 4;  ++m) s1 += Ri[m] * Rj[m];
        for (int m = 4; m < 9;  ++m) s2 += Ri[m] * Rj[m];
        for (int m = 9; m < 16; ++m) s3 += Ri[m] * Rj[m];
        op[23 + p] = s1; op[51 + p] = s2; op[79 + p] = s3;
    }

    for (int t = lane; t < 84; t += 32) {
        const float* Ri = R + T1l[t] * 16;
        const float* Rj = R + T2l[t] * 16;
        const float* Rk = R + T3l[t] * 16;
        float a211 = 0.f, a222 = 0.f, a312 = 0.f, a323 = 0.f;
        for (int n = 0; n < 3; ++n)
            for (int o = 0; o < 3; ++o) {
                float pre = Rj[1 + n] * Rk[1 + o];
                for (int m = 0; m < 5; ++m) a211 += W112[(n * 3 + o) * 5 + m] * Ri[4 + m] * pre;
            }
        for (int n = 0; n < 5; ++n)
            for (int o = 0; o < 5; ++o) {
                float pre = Rj[4 + n] * Rk[4 + o];
                for (int m = 0; m < 5; ++m) a222 += W222[(n * 5 + o) * 5 + m] * Ri[4 + m] * pre;
            }
        for (int n = 0; n < 3; ++n)
            for (int o = 0; o < 5; ++o) {
                float pre = Rj[1 + n] * Rk[4 + o];
                for (int m = 0; m < 7; ++m) a312 += W123[(n * 5 + o) * 7 + m] * Ri[9 + m] * pre;
            }
        for (int n = 0; n < 5; ++n)
            for (int o = 0; o < 7; ++o) {
                float pre = Rj[4 + n] * Rk[9 + o];
                for (int m = 0; m < 7; ++m) a323 += W233[(n * 7 + o) * 7 + m] * Ri[9 + m] * pre;
            }
        op[107 + t] = a211; op[191 + t] = a222; op[275 + t] = a312; op[359 + t] = a323;
    }
}

extern "C" void kernel_launch(void* const* d_in, const int* in_sizes, int n_in,
                              void* d_out, int out_size, void* d_ws, size_t ws_size,
                              hipStream_t stream) {
    (void)in_sizes; (void)n_in; (void)out_size; (void)ws_size;
    const int*   species = (const int*)d_in[0];
    const int*   esrc    = (const int*)d_in[1];
    const int*   edst    = (const int*)d_in[2];
    const float* dist    = (const float*)d_in[3];
    const float* vec     = (const float*)d_in[4];
    const float* sw      = (const float*)d_in[5];
    const float* table   = (const float*)d_in[6];
    const float* chem    = (const float*)d_in[7];
    float* out  = (float*)d_out;
    float* ws   = (float*)d_ws;
    int*   il   = (int*)ws + WS_IL;
    float* rhoi = ws + WS_RHOI;

    build_tables<<<1, 256, 0, stream>>>(table, chem, ws, il);
    int nR = NATOMS * NCHAN * 16;
    zero_kernel<<<(nR + 255) / 256, 256, 0, stream>>>(rhoi, nR);
    edge_kernel<<<(NEDGES + 255) / 256, 256, 0, stream>>>(esrc, edst, dist, vec, sw,
                                                          species, ws, rhoi);
    int waves = NATOMS / 16;                       // 1875
    onehot_wmma<<<(waves * 32 + 255) / 256, 256, 0, stream>>>(species, table, out);
    atom_kernel<<<(NATOMS + 7) / 8, 256, 0, stream>>>(rhoi, ws, il, out);
}